// CrossNetwork_35450660061553
// MI455X (gfx1250) — compile-verified
//
#include <hip/hip_runtime.h>
#include <cstdint>

// ---------------- problem constants ----------------
constexpr int Dk     = 1024;        // feature dim (fixed by reference)
constexpr int BLOCK  = 256;         // 8 waves (wave32)
constexpr int RPB    = 8;           // rows per block
constexpr int NWAVES = BLOCK / 32;

typedef float        f32x2  __attribute__((ext_vector_type(2)));
typedef float        f32x4  __attribute__((ext_vector_type(4)));
typedef float        f32x8  __attribute__((ext_vector_type(8)));
typedef unsigned int u32x4  __attribute__((ext_vector_type(4)));
typedef int          i32x4v __attribute__((ext_vector_type(4)));
typedef int          i32x8v __attribute__((ext_vector_type(8)));

#if defined(__HIP_DEVICE_COMPILE__) && \
    __has_builtin(__builtin_amdgcn_tensor_load_to_lds) && \
    __has_builtin(__builtin_amdgcn_s_wait_tensorcnt)
#define HAVE_TDM 1
#else
#define HAVE_TDM 0
#endif

#if defined(__HIP_DEVICE_COMPILE__) && \
    __has_builtin(__builtin_amdgcn_wmma_f32_16x16x4_f32)
#define HAVE_WMMA_RED 1
#else
#define HAVE_WMMA_RED 0
#endif

#if HAVE_TDM
// Build a minimal D# (1-D tensor, data_size = 4B, tile = nelem floats) and
// kick the Tensor Data Mover: global row -> LDS at lds_byte_off.
__device__ __forceinline__ void tdm_load_row(const float* gsrc,
                                             unsigned lds_byte_off,
                                             int nelem) {
  unsigned long long ga = (unsigned long long)(uintptr_t)gsrc;
  u32x4 g0;
  g0[0] = 1u;                                    // count=1 (valid), user mode
  g0[1] = lds_byte_off;                          // lds_addr [63:32]
  g0[2] = (unsigned)ga;                          // global_addr [95:64]
  g0[3] = (unsigned)((ga >> 32) & 0x01FFFFFFu)   // global_addr [120:96]
          | (2u << 30);                          // type=2 ("image")
  i32x8v g1;
  g1[0] = 2 << 16;                               // data_size=2 -> 4 bytes
  g1[1] = (int)((unsigned)(nelem & 0xFFFF) << 16);      // tensor_dim0 lo16
  g1[2] = ((nelem >> 16) & 0xFFFF) | (1 << 16);         // tensor_dim0 hi16, tensor_dim1=1
  g1[3] = (int)((unsigned)(nelem & 0xFFFF) << 16);      // tile_dim0
  g1[4] = 1;                                            // tile_dim1=1
  g1[5] = nelem;                                        // tensor_dim0_stride lo32
  g1[6] = 0;
  g1[7] = 0;
  i32x4v z4 = {0, 0, 0, 0};
#if __clang_major__ >= 23
  i32x8v z8 = {0, 0, 0, 0, 0, 0, 0, 0};
  __builtin_amdgcn_tensor_load_to_lds(g0, g1, z4, z4, z8, 0);
#else
  __builtin_amdgcn_tensor_load_to_lds(g0, g1, z4, z4, 0);
#endif
}

__device__ __forceinline__ unsigned lds_off(const void* p) {
  // Generic pointers to LDS carry the LDS byte offset in the low 32 bits.
  return (unsigned)(uintptr_t)p;
}
#endif  // HAVE_TDM

__global__ __launch_bounds__(BLOCK)
void cross_network_kernel(const float* __restrict__ in,
                          const float* __restrict__ W,
                          const float* __restrict__ bias,
                          float* __restrict__ out,
                          int L, int Btot) {
#if HAVE_TDM
  __shared__ __align__(16) float lds_w[2][Dk];
  __shared__ __align__(16) float lds_b[2][Dk];
#endif
#if HAVE_WMMA_RED
  // Transposed partials [buf][row (padded to 16)][wave]; rows 8-15 zeroed.
  __shared__ __align__(16) float lds_redT[2][16][NWAVES];
#else
  __shared__ __align__(16) float lds_red[2][NWAVES][RPB];
#endif

  const int t    = threadIdx.x;
  const int lane = t & 31;
  const int wid  = t >> 5;
  const int c0   = t * 4;                      // this thread's 4 columns
  const int row0 = (int)blockIdx.x * RPB;

#if HAVE_WMMA_RED
  ((float*)lds_redT)[t] = 0.0f;                // 2*16*8 == BLOCK floats
#endif

  // ---- load input tile: 8 rows x 4 cols per thread, registers only ----
  // Branch-free: clamp the row index (32-bit arithmetic; duplicate reads of
  // the last row are harmless, stores are guarded).
  f32x4 inp[RPB], x[RPB];
  const f32x4 zero4 = {0.0f, 0.0f, 0.0f, 0.0f};
#pragma unroll
  for (int r = 0; r < RPB; ++r) {
    int rr = row0 + r;
    rr = (rr < Btot) ? rr : (Btot - 1);
    const f32x4* p = (const f32x4*)(in + rr * Dk + c0);
    inp[r] = __builtin_nontemporal_load(p);    // streamed once, TH_NT
    x[r]   = inp[r];
  }

#if HAVE_TDM
  // Prologue: TDM-stage layer 0's W/b rows into LDS buffer 0.
  if (t == 0) {
    tdm_load_row(W,    lds_off(&lds_w[0][0]), Dk);
    tdm_load_row(bias, lds_off(&lds_b[0][0]), Dk);
  }
#endif

  for (int l = 0; l < L; ++l) {
    const int buf = l & 1;
#if HAVE_TDM
    if (t == 0) {
      if (l + 1 < L) {
        // Double-buffer: prefetch next layer while this one computes.
        tdm_load_row(W    + (l + 1) * Dk, lds_off(&lds_w[buf ^ 1][0]), Dk);
        tdm_load_row(bias + (l + 1) * Dk, lds_off(&lds_b[buf ^ 1][0]), Dk);
        __builtin_amdgcn_s_wait_tensorcnt(2);   // current-layer pair complete
      } else {
        __builtin_amdgcn_s_wait_tensorcnt(0);
      }
    }
    __syncthreads();                            // TDM data visible to all waves
    const f32x4 w4 = *(const f32x4*)&lds_w[buf][c0];
    const f32x4 b4 = *(const f32x4*)&lds_b[buf][c0];
#else
    const f32x4 w4 = *(const f32x4*)(W    + l * Dk + c0);
    const f32x4 b4 = *(const f32x4*)(bias + l * Dk + c0);
#endif

    // ---- per-row partial dot over this thread's 4 columns ----
    float p[RPB];
#pragma unroll
    for (int r = 0; r < RPB; ++r)
      p[r] = x[r][0] * w4[0] + x[r][1] * w4[1] + x[r][2] * w4[2] + x[r][3] * w4[3];

    // ---- wave32 butterfly reduction ----
#pragma unroll
    for (int off = 16; off > 0; off >>= 1) {
#pragma unroll
      for (int r = 0; r < RPB; ++r)
        p[r] += __shfl_xor(p[r], off, 32);
    }

    float rs[RPB];
#if HAVE_WMMA_RED
    // ---- cross-wave combine on the matrix unit ----
    // R[w][r] (8 waves x 8 rows) stored transposed; compute
    // D = A(16x4) x ones(4x16) twice with C-accumulate => D[i][j] = sum_w R[w][i].
    if (lane == 0) {
#pragma unroll
      for (int r = 0; r < RPB; ++r) lds_redT[buf][r][wid] = p[r];
    }
    __syncthreads();

    const int arow = lane & 15;
    const int acol = (lane >> 4) * 2;           // lanes 0-15: K0/K1, 16-31: K2/K3
    const f32x2 a0 = *(const f32x2*)&lds_redT[buf][arow][acol];
    const f32x2 a1 = *(const f32x2*)&lds_redT[buf][arow][acol + 4];
    const f32x2 onesB = {1.0f, 1.0f};
    f32x8 acc = {0.0f, 0.0f, 0.0f, 0.0f, 0.0f, 0.0f, 0.0f, 0.0f};
    acc = __builtin_amdgcn_wmma_f32_16x16x4_f32(false, a0, false, onesB,
                                                (short)0, acc, false, false);
    acc = __builtin_amdgcn_wmma_f32_16x16x4_f32(false, a1, false, onesB,
                                                (short)0, acc, false, false);
    // D rows 0-7 live on lanes 0-15 (VGPR r); broadcast to the upper half.
#pragma unroll
    for (int r = 0; r < RPB; ++r)
      rs[r] = __shfl(acc[r], lane & 15, 32);
#else
    // ---- fallback: cross-wave combine via LDS sums ----
    if (lane == 0) {
#pragma unroll
      for (int r = 0; r < RPB; ++r) lds_red[buf][wid][r] = p[r];
    }
    __syncthreads();

    f32x4 s0 = zero4, s1 = zero4;
#pragma unroll
    for (int w = 0; w < NWAVES; ++w) {
      s0 += *(const f32x4*)&lds_red[buf][w][0];
      s1 += *(const f32x4*)&lds_red[buf][w][4];
    }
#pragma unroll
    for (int r = 0; r < RPB; ++r) rs[r] = (r < 4) ? s0[r] : s1[r - 4];
#endif

    // ---- x = input * x1 + x + b ----
#pragma unroll
    for (int r = 0; r < RPB; ++r)
      x[r] = (x[r] + b4) + inp[r] * rs[r];
  }

  // ---- streamed writeback (guard only the tail) ----
#pragma unroll
  for (int r = 0; r < RPB; ++r) {
    if (row0 + r < Btot) {
      f32x4* o = (f32x4*)(out + (row0 + r) * Dk + c0);
      __builtin_nontemporal_store(x[r], o);
    }
  }
}

extern "C" void kernel_launch(void* const* d_in, const int* in_sizes, int n_in,
                              void* d_out, int out_size, void* d_ws, size_t ws_size,
                              hipStream_t stream) {
  const float* in   = (const float*)d_in[0];
  const float* W    = (const float*)d_in[1];
  const float* bias = (const float*)d_in[2];
  float* out        = (float*)d_out;

  const int Btot = in_sizes[0] / Dk;   // 16384
  const int L    = in_sizes[1] / Dk;   // 6
  const int grid = (Btot + RPB - 1) / RPB;

  hipLaunchKernelGGL(cross_network_kernel, dim3(grid), dim3(BLOCK), 0, stream,
                     in, W, bias, out, L, Btot);
}